// S_LSU_58291296141736
// MI455X (gfx1250) — compile-verified
//
#include <hip/hip_runtime.h>
#include <hip/hip_bf16.h>

#define HH 8
#define CC 64
#define TT 300
#define VV 25
#define II 16
#define NN 32
#define OO 64
#define TB 4
#define NCHUNK (TT / TB)   // 75
#define TILEW (TB * VV)    // 100

// padded LDS strides (chosen for 8B alignment + stride-4 bank rotation)
#define XSTR 68            // [col][c] tiles: 64 c + pad
#define KSTR 84            // [v][k] tiles in k_scores: 64 k + 16 overflow + pad
#define ASTR 28            // [w][k] A_h tile: 25 k -> 28
#define RSTR 28            // x~ tile row stride: 25 v -> 28

typedef float v2f __attribute__((ext_vector_type(2)));
typedef float v8f __attribute__((ext_vector_type(8)));

// D = A(16x4) x B(4x16) + C, exact fp32 (V_WMMA_F32_16X16X4_F32)
__device__ __forceinline__ v8f wmma_f32(float a0, float a1, float b0, float b1, v8f c) {
    v2f a; a[0] = a0; a[1] = a1;
    v2f b; b[0] = b0; b[1] = b1;
    return __builtin_amdgcn_wmma_f32_16x16x4_f32(false, a, false, b, (short)0, c, false, false);
}

__device__ __forceinline__ v2f lds2(const float* p) { return *(const v2f*)p; }

// ---- async global->LDS copy (GLOBAL_LOAD_ASYNC_TO_LDS_B32, ASYNCcnt) ------
#if defined(__HIP_DEVICE_COMPILE__) && __has_builtin(__builtin_amdgcn_global_load_async_to_lds_b32)
#define HAVE_ASYNC_LDS 1
#else
#define HAVE_ASYNC_LDS 0
#endif

#if HAVE_ASYNC_LDS
typedef __attribute__((address_space(1))) int gint_t;   // global
typedef __attribute__((address_space(3))) int lint_t;   // LDS
#endif

__device__ __forceinline__ void async_copy_b32(const float* g, float* l) {
#if HAVE_ASYNC_LDS
    __builtin_amdgcn_global_load_async_to_lds_b32((gint_t*)g, (lint_t*)l, 0, 0);
#else
    *l = *g;
#endif
}

__device__ __forceinline__ void async_wait_all() {
#if HAVE_ASYNC_LDS
#if __has_builtin(__builtin_amdgcn_s_wait_asynccnt)
    __builtin_amdgcn_s_wait_asynccnt(0);
#else
    asm volatile("s_wait_asynccnt 0x0" ::: "memory");
#endif
#endif
}

// ---------------------------------------------------------------------------
// Kernel 1: attention scores  S[h,n,v,w] = (1/4800) * sum_{i,t} a[i,t,v]*b[i,t,w]
//   stage 1 (WMMA): a/b = wa/wb[h] (16x64) @ x-tile (64x100) + bias
//   stage 2 (WMMA): S quadrants (16x16) accumulate over K=(i,tl)=64 per chunk
// ---------------------------------------------------------------------------
__global__ __launch_bounds__(128) void k_scores(
    const float* __restrict__ x, const float* __restrict__ wa, const float* __restrict__ ba,
    const float* __restrict__ wb, const float* __restrict__ bb, float* __restrict__ S)
{
    __shared__ float xtT[112 * XSTR];  // [col=(tl*25+v)][c], rows 100..111 zero
    __shared__ float asT[32 * KSTR];   // [v][k= tl*16+i], k>=64 is spill pad
    __shared__ float bsT[32 * KSTR];

    const int tid  = threadIdx.x;
    const int lane = tid & 31;
    const int wid  = tid >> 5;
    const int lm   = lane & 15;
    const int half = (lane < 16) ? 0 : 2;   // fragment k offset
    const int mh   = (lane < 16) ? 0 : 8;   // C/D row offset

    const int h = blockIdx.x % HH;
    const int n = blockIdx.x / HH;          // consecutive blocks share x[n] in L2

    // hoisted weight fragments (A operand: M=i, K=c) and biases
    float fa0[16], fa1[16], fb0[16], fb1[16];
#pragma unroll
    for (int ks = 0; ks < 16; ++ks) {
        const int k0 = ks * 4 + half;
        fa0[ks] = wa[(h * II + lm) * CC + k0];
        fa1[ks] = wa[(h * II + lm) * CC + k0 + 1];
        fb0[ks] = wb[(h * II + lm) * CC + k0];
        fb1[ks] = wb[(h * II + lm) * CC + k0 + 1];
    }
    float baf[8], bbf[8];
#pragma unroll
    for (int r = 0; r < 8; ++r) { baf[r] = ba[h * II + r + mh]; bbf[r] = bb[h * II + r + mh]; }

    // zero pads (written once; main loads never touch them)
    for (int i = tid; i < 12 * XSTR; i += 128) xtT[(100 + i / XSTR) * XSTR + (i % XSTR)] = 0.f;
    for (int i = tid; i < 7 * KSTR; i += 128) {
        asT[(VV + i / KSTR) * KSTR + (i % KSTR)] = 0.f;
        bsT[(VV + i / KSTR) * KSTR + (i % KSTR)] = 0.f;
    }

    // S quadrant owned by this wave
    const int smt = wid >> 1, snt = wid & 1;
    v8f sacc;
#pragma unroll
    for (int r = 0; r < 8; ++r) sacc[r] = 0.f;

    const float* xbase = x + ((size_t)n * CC) * TT * VV;

    for (int tc = 0; tc < NCHUNK; ++tc) {
        // load x[n,:,4 t's,:] as float4, scatter transposed into LDS
        const float* xsrc = xbase + (size_t)tc * TILEW;
        for (int i = tid; i < CC * TILEW / 4; i += 128) {
            const int c = i / 25, g = (i % 25) * 4;
            const float4 xv = *(const float4*)(xsrc + (size_t)c * TT * VV + g);
            xtT[(g + 0) * XSTR + c] = xv.x;
            xtT[(g + 1) * XSTR + c] = xv.y;
            xtT[(g + 2) * XSTR + c] = xv.z;
            xtT[(g + 3) * XSTR + c] = xv.w;
        }
        // prefetch next chunk's cachelines (global_prefetch_b8) to hide latency
        if (tc + 1 < NCHUNK) {
            const float* xnxt = xbase + (size_t)(tc + 1) * TILEW;
            const int li = tid * 2;
            __builtin_prefetch(xnxt + (size_t)(li >> 2) * (TT * VV) + (li & 3) * 32, 0, 0);
            __builtin_prefetch(xnxt + (size_t)((li + 1) >> 2) * (TT * VV) + ((li + 1) & 3) * 32, 0, 0);
        }
        __syncthreads();

        // ---- projections: M=16(i), K=64(c), N=112 (7 N-tiles over 4 waves) ----
        for (int pass = 0; pass < 2; ++pass) {
            const int nt = wid + pass * 4;
            if (nt >= 7) break;
            const int col = nt * 16 + lm;
            v8f accA, accB;
#pragma unroll
            for (int r = 0; r < 8; ++r) { accA[r] = baf[r]; accB[r] = bbf[r]; }
#pragma unroll
            for (int ks = 0; ks < 16; ++ks) {
                const int k0 = ks * 4 + half;
                const v2f bf = lds2(&xtT[col * XSTR + k0]);       // one ds_load_b64
                accA = wmma_f32(fa0[ks], fa1[ks], bf[0], bf[1], accA);
                accB = wmma_f32(fb0[ks], fb1[ks], bf[0], bf[1], accB);
            }
            // store transposed [v][k=tl*16+i]; col>=100 lands in k>=64 spill pad
            const int stl = col / VV, sv = col % VV;
#pragma unroll
            for (int r = 0; r < 8; ++r) {
                const int i_ = r + mh;
                asT[sv * KSTR + stl * 16 + i_] = accA[r];
                bsT[sv * KSTR + stl * 16 + i_] = accB[r];
            }
        }
        __syncthreads();

        // ---- S quadrant: D[v,w] += sum_k a[v,k]*b[k,w], K=64 this chunk ----
#pragma unroll
        for (int ks = 0; ks < 16; ++ks) {
            const int k0 = ks * 4 + half;
            const v2f af = lds2(&asT[(smt * 16 + lm) * KSTR + k0]);
            const v2f bf = lds2(&bsT[(snt * 16 + lm) * KSTR + k0]);
            sacc = wmma_f32(af[0], af[1], bf[0], bf[1], sacc);
        }
        __syncthreads();
    }

    // store this wave's S quadrant
    float* sdst = S + ((size_t)(h * NN + n)) * VV * VV;
#pragma unroll
    for (int r = 0; r < 8; ++r) {
        const int v = smt * 16 + r + mh;
        const int w = snt * 16 + lm;
        if (v < VV && w < VV) sdst[v * VV + w] = sacc[r] * (1.0f / (II * TT));
    }
}

// ---------------------------------------------------------------------------
// Kernel 2: A = softmax(S, axis=v) + DepM   (one lane per column w)
// ---------------------------------------------------------------------------
__global__ __launch_bounds__(32) void k_softmax(
    const float* __restrict__ S, const float* __restrict__ DepM, float* __restrict__ A)
{
    const int h = blockIdx.x % HH, n = blockIdx.x / HH;
    const int w = threadIdx.x;
    if (w >= VV) return;
    const float* s = S + ((size_t)(h * NN + n)) * VV * VV;
    float col[VV];
    float mx = -3.4e38f;
#pragma unroll
    for (int v = 0; v < VV; ++v) { col[v] = s[v * VV + w]; mx = fmaxf(mx, col[v]); }
    float sum = 0.f;
#pragma unroll
    for (int v = 0; v < VV; ++v) { col[v] = __expf(col[v] - mx); sum += col[v]; }
    const float inv = 1.0f / sum;
    float* a = A + ((size_t)(h * NN + n)) * VV * VV;
#pragma unroll
    for (int v = 0; v < VV; ++v)
        a[v * VV + w] = col[v] * inv + DepM[(h * VV + v) * VV + w];
}

__global__ void k_zero_stats(float* __restrict__ stats)
{
    if (threadIdx.x < 2 * OO) stats[threadIdx.x] = 0.f;
}

// ---------------------------------------------------------------------------
// Kernel 3: y[n,o,t,w] = sum_h wd[h] @ ( x~[n] @ A[h,n] ) + sum_h bd ; BN stats
//   GEMM1: x~-tile (256 x 28pad) @ A_h (28 x 32pad) -> z (LDS, transposed)
//   GEMM2: wd[h] (64 x 64) @ z (64 x 112pad)        -> register accumulators
//   A_h is gathered per head with async global->LDS copies (ASYNCcnt)
// ---------------------------------------------------------------------------
__global__ __launch_bounds__(128) void k_out(
    const float* __restrict__ x, const float* __restrict__ A, const float* __restrict__ wd,
    const float* __restrict__ bd, float* __restrict__ y, float* __restrict__ stats)
{
    __shared__ float xt2[256 * RSTR];  // [row=(c*4+tl)][v], v 25..27 zero
    __shared__ float AhT[32 * ASTR];   // [w][k], zero padded (pad written once)
    __shared__ float ztT[112 * XSTR];  // [col=(tl*25+w)][c], rows 100..111 zero
    __shared__ float bsum[OO];
    __shared__ float bsq[OO];

    const int tid  = threadIdx.x;
    const int lane = tid & 31;
    const int wid  = tid >> 5;
    const int lm   = lane & 15;
    const int half = (lane < 16) ? 0 : 2;
    const int mh   = (lane < 16) ? 0 : 8;

    const int n  = blockIdx.x / NCHUNK;
    const int tb = blockIdx.x % NCHUNK;

    // load x tile as float4, scatter into padded row layout (addr = row*28 + v)
    const float* xsrc = x + ((size_t)n * CC) * TT * VV + (size_t)tb * TILEW;
    for (int i = tid; i < CC * TILEW / 4; i += 128) {
        const int c = i / 25, g = (i % 25) * 4;
        const float4 xv = *(const float4*)(xsrc + (size_t)c * TT * VV + g);
#pragma unroll
        for (int e = 0; e < 4; ++e) {
            const int q = g + e;
            const float val = (e == 0) ? xv.x : (e == 1) ? xv.y : (e == 2) ? xv.z : xv.w;
            xt2[(c * TB + q / VV) * RSTR + (q % VV)] = val;
        }
    }
    for (int i = tid; i < 256 * 3; i += 128) xt2[(i / 3) * RSTR + VV + (i % 3)] = 0.f;
    for (int i = tid; i < 12 * XSTR; i += 128) ztT[(100 + i / XSTR) * XSTR + (i % XSTR)] = 0.f;
    for (int i = tid; i < 32 * ASTR; i += 128) AhT[i] = 0.f;   // pad stays 0 forever
    if (tid < OO) { bsum[tid] = 0.f; bsq[tid] = 0.f; }

    // accumulators: wave owns o in [wid*16, wid*16+16), 7 col-tiles of 16
    float bsv[8];
#pragma unroll
    for (int r = 0; r < 8; ++r) {
        const int o = wid * 16 + r + mh;
        float s = 0.f;
#pragma unroll
        for (int h = 0; h < HH; ++h) s += bd[h * OO + o];
        bsv[r] = s;
    }
    v8f acc[7];
#pragma unroll
    for (int nt = 0; nt < 7; ++nt)
#pragma unroll
        for (int r = 0; r < 8; ++r) acc[nt][r] = bsv[r];

    for (int h = 0; h < HH; ++h) {
        __syncthreads();   // xt2/pads ready (iter 0); prev GEMM2 done (iter >0)
        // gather A_h transposed into LDS via async copies (overlapped, then waited)
        const float* Asrc = A + ((size_t)(h * NN + n)) * VV * VV;
#pragma unroll
        for (int j = 0; j < 5; ++j) {
            const int i = tid + j * 128;
            if (i < VV * VV) {
                const int w = i / VV, k = i % VV;
                async_copy_b32(&Asrc[k * VV + w], &AhT[w * ASTR + k]);
            }
        }
        async_wait_all();
        __syncthreads();

        // ---- GEMM1: z[row, w] = sum_v x~[row,v] * A_h[v,w] ----
        v2f bA0[7], bA1[7];               // A_h fragments, reused by 4 M-tiles
#pragma unroll
        for (int ks = 0; ks < 7; ++ks) {
            const int k0 = ks * 4 + half;
            bA0[ks] = lds2(&AhT[lm * ASTR + k0]);
            bA1[ks] = lds2(&AhT[(16 + lm) * ASTR + k0]);
        }
#pragma unroll
        for (int mi = 0; mi < 4; ++mi) {
            const int mt  = wid + mi * 4;
            const int row = mt * 16 + lm;
            v8f z0, z1;
#pragma unroll
            for (int r = 0; r < 8; ++r) { z0[r] = 0.f; z1[r] = 0.f; }
#pragma unroll
            for (int ks = 0; ks < 7; ++ks) {
                const int k0 = ks * 4 + half;
                const v2f af = lds2(&xt2[row * RSTR + k0]);       // one ds_load_b64
                z0 = wmma_f32(af[0], af[1], bA0[ks][0], bA0[ks][1], z0);
                z1 = wmma_f32(af[0], af[1], bA1[ks][0], bA1[ks][1], z1);
            }
#pragma unroll
            for (int r = 0; r < 8; ++r) {
                const int m  = mt * 16 + r + mh;
                const int zc = m >> 2, ztl = m & 3;
                if (lm < VV)      ztT[(ztl * VV + lm) * XSTR + zc]      = z0[r];
                if (16 + lm < VV) ztT[(ztl * VV + 16 + lm) * XSTR + zc] = z1[r];
            }
        }
        __syncthreads();

        // ---- GEMM2: M=o (wave tile), K=c=64, N=(tl,w)=112 ----
        float wdf0[16], wdf1[16];
        const int orow = wid * 16 + lm;
#pragma unroll
        for (int ks = 0; ks < 16; ++ks) {
            const int k0 = ks * 4 + half;
            wdf0[ks] = wd[((size_t)h * OO + orow) * CC + k0];
            wdf1[ks] = wd[((size_t)h * OO + orow) * CC + k0 + 1];
        }
#pragma unroll
        for (int nt = 0; nt < 7; ++nt) {
            const int col = nt * 16 + lm;
#pragma unroll
            for (int ks = 0; ks < 16; ++ks) {
                const int k0 = ks * 4 + half;
                const v2f bf = lds2(&ztT[col * XSTR + k0]);        // one ds_load_b64
                acc[nt] = wmma_f32(wdf0[ks], wdf1[ks], bf[0], bf[1], acc[nt]);
            }
        }
    }

    __syncthreads();
    // epilogue: store y, reduce BN statistics
#pragma unroll
    for (int nt = 0; nt < 7; ++nt) {
        const int col = nt * 16 + lm;
        if (col < TILEW) {
            const int tl = col / VV, w = col % VV;
#pragma unroll
            for (int r = 0; r < 8; ++r) {
                const int o = wid * 16 + r + mh;
                const float val = acc[nt][r];
                y[(((size_t)n * OO + o) * TT + tb * TB + tl) * VV + w] = val;
                atomicAdd(&bsum[o], val);
                atomicAdd(&bsq[o], val * val);
            }
        }
    }
    __syncthreads();
    if (tid < OO) {
        atomicAdd(&stats[tid],      bsum[tid]);
        atomicAdd(&stats[OO + tid], bsq[tid]);
    }
}

// ---------------------------------------------------------------------------
// Kernel 4: BatchNorm (batch stats, biased var) + residual + ReLU, in place
// ---------------------------------------------------------------------------
__global__ __launch_bounds__(256) void k_bn(
    const float* __restrict__ x, const float* __restrict__ stats,
    const float* __restrict__ gamma, const float* __restrict__ beta,
    float* __restrict__ out)
{
    const size_t total = (size_t)NN * OO * TT * VV;
    const size_t idx = (size_t)blockIdx.x * 256 + threadIdx.x;
    if (idx >= total) return;
    const int o = (int)((idx / (TT * VV)) % OO);
    const float Mcnt = (float)((size_t)NN * TT * VV);
    const float mean = stats[o] / Mcnt;
    const float var  = stats[OO + o] / Mcnt - mean * mean;
    const float inv  = rsqrtf(var + 1e-5f);
    const float r    = (out[idx] - mean) * inv * gamma[o] + beta[o] + x[idx];
    out[idx] = fmaxf(r, 0.f);
}

extern "C" void kernel_launch(void* const* d_in, const int* in_sizes, int n_in,
                              void* d_out, int out_size, void* d_ws, size_t ws_size,
                              hipStream_t stream)
{
    const float* x     = (const float*)d_in[0];
    const float* DepM  = (const float*)d_in[1];
    const float* wa    = (const float*)d_in[2];
    const float* ba    = (const float*)d_in[3];
    const float* wb    = (const float*)d_in[4];
    const float* bb    = (const float*)d_in[5];
    const float* wd    = (const float*)d_in[6];
    const float* bd    = (const float*)d_in[7];
    const float* gamma = (const float*)d_in[8];
    const float* beta  = (const float*)d_in[9];
    float* out = (float*)d_out;

    float* S     = (float*)d_ws;                  // [H,N,25,25]
    float* A     = S + (size_t)HH * NN * VV * VV; // [H,N,25,25]
    float* stats = A + (size_t)HH * NN * VV * VV; // [2*64] sum / sumsq

    k_scores  <<<HH * NN, 128, 0, stream>>>(x, wa, ba, wb, bb, S);
    k_softmax <<<HH * NN,  32, 0, stream>>>(S, DepM, A);
    k_zero_stats<<<1, 128, 0, stream>>>(stats);
    k_out     <<<NN * NCHUNK, 128, 0, stream>>>(x, A, wd, bd, out, stats);
    const size_t total = (size_t)NN * OO * TT * VV;
    k_bn      <<<(int)((total + 255) / 256), 256, 0, stream>>>(x, stats, gamma, beta, out);
}